// MemoryModule_20959440405248
// MI455X (gfx1250) — compile-verified
//
#include <hip/hip_runtime.h>
#include <hip/hip_bf16.h>
#include <math.h>

// Problem constants (from reference setup_inputs)
#define BATCH   4
#define DFEAT   32768            // c*h*w = 128*16*16
#define NMEM    2000
#define NTILES  (NMEM / 16)      // 125
#define SEG     8                // d-range split per (tile,b)
#define DSEG    (DFEAT / SEG)    // 4096
#define OUTSZ   (BATCH * DFEAT)  // 131072
#define KMAX    32
#define EPS     1e-8f

typedef __attribute__((ext_vector_type(2))) float v2f;
typedef __attribute__((ext_vector_type(8))) float v8f;

// ---------------------------------------------------------------------------
// Phase 0: zero the dots/msq accumulators in workspace
// ---------------------------------------------------------------------------
__global__ void zero_kernel(float* __restrict__ p, int n) {
    int i = blockIdx.x * blockDim.x + threadIdx.x;
    if (i < n) p[i] = 0.0f;
}

// ---------------------------------------------------------------------------
// Phase 1: per-(n,b) dot(q,m) via V_WMMA_F32_16X16X4_F32 and ||m||^2 via VALU.
// One wave (32 threads) per block. blockIdx.x encodes (tile, b, seg).
//
// A (16x4 fp32): lane L (M = L&15), half = L>>4: VGPR0 = m[M][d+2*half],
//                VGPR1 = m[M][d+2*half+1]   (documented 32-bit A layout)
// B (4x16 fp32): replicated q chunk across all 16 columns, same K mapping as A,
//                so every column of C holds the same dots -> column layout of B
//                cannot break correctness, only K placement matters.
// C (16x16 fp32): VGPR v: lanes 0-15 -> M=v, lanes 16-31 -> M=v+8.
// ---------------------------------------------------------------------------
__global__ __launch_bounds__(32) void phase1_dots(
    const float* __restrict__ q,     // [BATCH, DFEAT]
    const float* __restrict__ mem,   // [NMEM, BATCH, DFEAT]
    float* __restrict__ dots,        // [NMEM, BATCH]
    float* __restrict__ msq)         // [NMEM, BATCH]
{
    const int lane = threadIdx.x;
    const int half = lane >> 4;        // 0: K=0,1  1: K=2,3
    const int rowA = lane & 15;        // A-matrix row (= n within tile)

    int id = blockIdx.x;
    const int seg  = id % SEG;  id /= SEG;
    const int b    = id % BATCH; id /= BATCH;
    const int tile = id;                       // 0..NTILES-1

    const int     n     = tile * 16 + rowA;
    const size_t  mbase = ((size_t)n * BATCH + b) * DFEAT;
    const size_t  qbase = (size_t)b * DFEAT;
    const int     d0    = seg * DSEG + 2 * half;

    v8f c0 = {}, c1 = {}, c2 = {}, c3 = {};
    float ss = 0.0f;

    const int iters = DSEG / 16;   // 4 K-chunks of 4 per iteration
    for (int it = 0; it < iters; ++it) {
        const int d = d0 + it * 16;

        v2f a0 = *(const v2f*)(mem + mbase + d);
        v2f a1 = *(const v2f*)(mem + mbase + d + 4);
        v2f a2 = *(const v2f*)(mem + mbase + d + 8);
        v2f a3 = *(const v2f*)(mem + mbase + d + 12);

        v2f b0 = *(const v2f*)(q + qbase + d);
        v2f b1 = *(const v2f*)(q + qbase + d + 4);
        v2f b2 = *(const v2f*)(q + qbase + d + 8);
        v2f b3 = *(const v2f*)(q + qbase + d + 12);

        // 4 independent accumulator chains -> WMMAs can pipeline
        c0 = __builtin_amdgcn_wmma_f32_16x16x4_f32(false, a0, false, b0,
                                                   (short)0, c0, false, false);
        c1 = __builtin_amdgcn_wmma_f32_16x16x4_f32(false, a1, false, b1,
                                                   (short)0, c1, false, false);
        c2 = __builtin_amdgcn_wmma_f32_16x16x4_f32(false, a2, false, b2,
                                                   (short)0, c2, false, false);
        c3 = __builtin_amdgcn_wmma_f32_16x16x4_f32(false, a3, false, b3,
                                                   (short)0, c3, false, false);

        ss = fmaf(a0.x, a0.x, ss); ss = fmaf(a0.y, a0.y, ss);
        ss = fmaf(a1.x, a1.x, ss); ss = fmaf(a1.y, a1.y, ss);
        ss = fmaf(a2.x, a2.x, ss); ss = fmaf(a2.y, a2.y, ss);
        ss = fmaf(a3.x, a3.x, ss); ss = fmaf(a3.y, a3.y, ss);
    }

    v8f cs = c0 + c1 + c2 + c3;

    // ||m||^2: lane L covers K%4 in {2h,2h+1}; pair with lane L^16 for full row
    ss += __shfl_xor(ss, 16);
    if (lane < 16)
        atomicAdd(&msq[(size_t)n * BATCH + b], ss);

    // dots: C layout -> lane 0 holds rows 0..7, lane 16 holds rows 8..15
    if (rowA == 0) {
        #pragma unroll
        for (int v = 0; v < 8; ++v) {
            const int nrow = tile * 16 + half * 8 + v;
            atomicAdd(&dots[(size_t)nrow * BATCH + b], cs[v]);
        }
    }
}

// ---------------------------------------------------------------------------
// Phase 2 (single block): query norms, sims, top-k, softmax
// ---------------------------------------------------------------------------
__global__ __launch_bounds__(256) void phase2_topk(
    const float* __restrict__ q,
    const float* __restrict__ dots,
    const float* __restrict__ msq,
    const int*   __restrict__ kptr,
    float* __restrict__ weights,     // [KMAX]
    int*   __restrict__ topidx)      // [KMAX]
{
    __shared__ float red[256];
    __shared__ int   redi[256];
    __shared__ float qn[BATCH];
    __shared__ float sims[NMEM];
    __shared__ float tv[KMAX];
    __shared__ int   ti[KMAX];

    const int t = threadIdx.x;
    const int k = min(*kptr, KMAX);

    // query norms per batch
    for (int b = 0; b < BATCH; ++b) {
        float p = 0.0f;
        for (int i = t; i < DFEAT; i += 256) {
            float v = q[(size_t)b * DFEAT + i];
            p = fmaf(v, v, p);
        }
        red[t] = p;
        __syncthreads();
        for (int s = 128; s > 0; s >>= 1) {
            if (t < s) red[t] += red[t + s];
            __syncthreads();
        }
        if (t == 0) qn[b] = fmaxf(sqrtf(red[0]), EPS);
        __syncthreads();
    }

    // cosine sims, mean over batch
    for (int n = t; n < NMEM; n += 256) {
        float s = 0.0f;
        for (int b = 0; b < BATCH; ++b) {
            float mn = fmaxf(sqrtf(msq[(size_t)n * BATCH + b]), EPS);
            s += dots[(size_t)n * BATCH + b] / (mn * qn[b]);
        }
        sims[n] = s * (1.0f / BATCH);
    }
    __syncthreads();

    // iterative top-k (ties -> lower index, matches lax.top_k ordering intent)
    for (int kk = 0; kk < k; ++kk) {
        float bv = -1e30f;
        int   bi = 0x7fffffff;
        for (int n = t; n < NMEM; n += 256) {
            float v = sims[n];
            if (v > bv || (v == bv && n < bi)) { bv = v; bi = n; }
        }
        red[t] = bv; redi[t] = bi;
        __syncthreads();
        for (int s = 128; s > 0; s >>= 1) {
            if (t < s) {
                if (red[t + s] > red[t] ||
                    (red[t + s] == red[t] && redi[t + s] < redi[t])) {
                    red[t]  = red[t + s];
                    redi[t] = redi[t + s];
                }
            }
            __syncthreads();
        }
        if (t == 0) {
            tv[kk] = red[0];
            ti[kk] = redi[0];
            sims[redi[0]] = -1e30f;   // remove from future rounds
        }
        __syncthreads();
    }

    // softmax over the k selected values
    if (t == 0) {
        float m = tv[0];
        for (int i = 1; i < k; ++i) m = fmaxf(m, tv[i]);
        float sum = 0.0f;
        for (int i = 0; i < k; ++i) {
            float e = expf(tv[i] - m);
            weights[i] = e;
            sum += e;
        }
        float inv = 1.0f / sum;
        for (int i = 0; i < k; ++i) {
            weights[i] *= inv;
            topidx[i]   = ti[i];
        }
    }
}

// ---------------------------------------------------------------------------
// Phase 3: weighted gather-sum of the k selected memory items
// ---------------------------------------------------------------------------
__global__ __launch_bounds__(256) void phase3_wsum(
    const float* __restrict__ mem,
    const float* __restrict__ weights,
    const int*   __restrict__ topidx,
    const int*   __restrict__ kptr,
    float* __restrict__ out)
{
    const int j = blockIdx.x * 256 + threadIdx.x;
    if (j >= OUTSZ) return;
    const int k = min(*kptr, KMAX);
    float acc = 0.0f;
    for (int i = 0; i < k; ++i) {
        acc = fmaf(weights[i], mem[(size_t)topidx[i] * OUTSZ + j], acc);
    }
    out[j] = acc;
}

// ---------------------------------------------------------------------------
extern "C" void kernel_launch(void* const* d_in, const int* in_sizes, int n_in,
                              void* d_out, int out_size, void* d_ws, size_t ws_size,
                              hipStream_t stream) {
    const float* q    = (const float*)d_in[0];   // [4,128,16,16]
    const float* mem  = (const float*)d_in[1];   // [2000,4,128,16,16]
    const int*   kptr = (const int*)d_in[2];     // scalar k on device

    float* ws      = (float*)d_ws;
    float* dots    = ws;                 // NMEM*BATCH = 8000 floats
    float* msq     = ws + NMEM * BATCH;  // 8000 floats
    float* weights = ws + 2 * NMEM * BATCH;            // KMAX floats
    int*   topidx  = (int*)(ws + 2 * NMEM * BATCH + KMAX);

    const int accn = 2 * NMEM * BATCH;   // 16000
    zero_kernel<<<(accn + 255) / 256, 256, 0, stream>>>(ws, accn);

    phase1_dots<<<NTILES * BATCH * SEG, 32, 0, stream>>>(q, mem, dots, msq);

    phase2_topk<<<1, 256, 0, stream>>>(q, dots, msq, kptr, weights, topidx);

    phase3_wsum<<<(OUTSZ + 255) / 256, 256, 0, stream>>>(mem, weights, topidx,
                                                         kptr, (float*)d_out);
}